// TolmanEichenbaumMachine_44023414784316
// MI455X (gfx1250) — compile-verified
//
#include <hip/hip_runtime.h>
#include <hip/hip_bf16.h>
#include <stddef.h>

// ---- problem constants (match reference) ----
#define NSTREAMS 5
#define SDIM     10
#define GTOT     120
#define GDTOT    40
#define PDIM     400
#define STOT     50
#define NSENS    45
#define TSTEPS   75
#define BSIZE    32
#define MEMITS   5
#define HDIM     200      // 20*S_DIM
#define NTHREADS 1024

// gdown index: streams start at 0,30,60,84,102 with down dims 10,10,8,6,6
__constant__ int c_gdown[GDTOT] = {
    0,1,2,3,4,5,6,7,8,9,
    30,31,32,33,34,35,36,37,38,39,
    60,61,62,63,64,65,66,67,
    84,85,86,87,88,89,
    102,103,104,105,106,107};

__constant__ float c_alpha[NSTREAMS] = {0.01f, 0.7f, 0.91f, 0.97f, 0.99f};

typedef __attribute__((ext_vector_type(2))) float v2f;
typedef __attribute__((ext_vector_type(8))) float v8f;

__device__ __forceinline__ float leaky_clip(float x) {
    x = fminf(1.0f, fmaxf(-1.0f, x));
    return x >= 0.0f ? x : 0.01f * x;
}

__launch_bounds__(NTHREADS, 1)
__global__ void tem_scan_kernel(
    const int*   __restrict__ x_idx,    // (B,T)
    const int*   __restrict__ actions,  // (B,T)
    const float* __restrict__ two_hot,  // (45,10)
    const float* __restrict__ W_trans,  // (4,120,120)
    const float* __restrict__ W_sig_p,  // (120,120)
    const float* __restrict__ b_sig_p,  // (120)
    const float* __restrict__ W_up,     // (40,120)
    const float* __restrict__ W_sig_m,  // (120,120)
    const float* __restrict__ b_sig_m,  // (120)
    const float* __restrict__ W_repeat, // (400,40)
    const float* __restrict__ W_tile,   // (400,50)
    const float* __restrict__ w_x,      // (1)
    const float* __restrict__ b_x,      // (10)
    const float* __restrict__ W1,       // (10,200)
    const float* __restrict__ b1,       // (200)
    const float* __restrict__ W2,       // (200,45)
    const float* __restrict__ b2,       // (45)
    float*       __restrict__ out,      // (T,B,45)
    float*       __restrict__ Mws)      // (B,400,400) fp32 workspace, L2-resident
{
    const int b    = blockIdx.x;
    const int tid  = threadIdx.x;
    const int lane = tid & 31;
    const int wave = tid >> 5;
    float* Mb = Mws + (size_t)b * (PDIM * PDIM);

    // Attractor state: H is 400x16; col0 = p_x chain, col1 = p_gen chain, cols 2..15 stay zero
    __shared__ float Hb[PDIM * 16];           // 25.6 KB
    __shared__ float Yb[2][PDIM];
    __shared__ float xf[STOT];
    __shared__ float g_vec[GTOT];
    __shared__ float g_prior_mu[GTOT];
    __shared__ float g_prior_lv[GTOT];
    __shared__ float g_mem_mu[GTOT];
    __shared__ float px_raw[PDIM];
    __shared__ float tmp40[GDTOT];
    __shared__ float gsel_prior[GDTOT];
    __shared__ float gsel_new[GDTOT];
    __shared__ float dvec[PDIM];
    __shared__ float svec[PDIM];
    __shared__ float x_dec[SDIM];
    __shared__ float hid[HDIM];

    // ---- init carry state (M0 = 0, g0 = 0, xf0 = 0) ----
    for (int i = tid; i < PDIM * PDIM; i += NTHREADS) Mb[i] = 0.0f;
    for (int i = tid; i < PDIM * 16;   i += NTHREADS) Hb[i] = 0.0f;
    if (tid < STOT) xf[tid]   = 0.0f;
    if (tid < GTOT) g_vec[tid] = 0.0f;
    __threadfence_block();
    __syncthreads();

    const float wx = w_x[0];

    for (int t = 0; t < TSTEPS; ++t) {
        const int xi = x_idx[b * TSTEPS + t];
        const int ai = actions[b * TSTEPS + t];

        // xf = alpha*xf + (1-alpha)*tile(two_hot[xi])
        if (tid < STOT) {
            float a  = c_alpha[tid / SDIM];
            float xs = two_hot[xi * SDIM + (tid % SDIM)];
            xf[tid] = a * xf[tid] + (1.0f - a) * xs;
        }
        // g_prior_mu = leaky(clip(g @ W_trans[ai]))
        if (tid < GTOT) {
            const float* W = W_trans + (size_t)ai * GTOT * GTOT;
            float acc = 0.0f;
            for (int gi = 0; gi < GTOT; ++gi) acc += g_vec[gi] * W[gi * GTOT + tid];
            g_prior_mu[tid] = leaky_clip(acc);
        }
        __syncthreads();

        // g_prior_lv = scale_log_sigma(g_prior_mu @ W_sig_p + b_sig_p)
        if (tid < GTOT) {
            float acc = b_sig_p[tid];
            for (int i = 0; i < GTOT; ++i) acc += g_prior_mu[i] * W_sig_p[i * GTOT + tid];
            g_prior_lv[tid] = -2.0f + 6.0f * tanhf(acc * (1.0f / 6.0f));
        }
        if (tid < GDTOT) gsel_prior[tid] = g_prior_mu[c_gdown[tid]];
        // px_raw = xf @ W_tile^T   (also the attractor-0 init and the p_inf gate)
        if (tid < PDIM) {
            const float* wr = W_tile + tid * STOT;
            float acc = 0.0f;
            for (int s = 0; s < STOT; ++s) acc += xf[s] * wr[s];
            px_raw[tid] = acc;
        }
        __syncthreads();

        // attractor inits: col0 = px_raw, col1 = g_prior_mu[gdown] @ W_repeat^T
        if (tid < PDIM) {
            const float* wr = W_repeat + tid * GDTOT;
            float acc = 0.0f;
            for (int d = 0; d < GDTOT; ++d) acc += gsel_prior[d] * wr[d];
            Hb[tid * 16 + 0] = px_raw[tid];
            Hb[tid * 16 + 1] = acc;
        }
        __syncthreads();

        // ---- joint attractor: 5 iterations of Y = M @ H via V_WMMA_F32_16X16X4_F32 ----
        const int n     = lane & 15;      // N (and A-row) index
        const int khalf = lane >> 4;      // 0 -> K{0,1}, 1 -> K{2,3}
        for (int it = 0; it < MEMITS; ++it) {
            for (int tile = wave; tile < PDIM / 16; tile += NTHREADS / 32) {
                const int p0 = tile * 16;
                v8f c = {0.f,0.f,0.f,0.f,0.f,0.f,0.f,0.f};
                const float* Arow = Mb + (size_t)(p0 + n) * PDIM + 2 * khalf;
                for (int k4 = 0; k4 < PDIM; k4 += 4) {
                    // A 16x4 fp32: lane n holds M[p0+n][k4 + 2*khalf + {0,1}]  (8B aligned)
                    v2f a = *(const v2f*)(Arow + k4);
                    // B 4x16 fp32: lane holds H[k4 + 2*khalf + {0,1}][n]
                    v2f bb;
                    bb.x = Hb[(k4 + 2 * khalf + 0) * 16 + n];
                    bb.y = Hb[(k4 + 2 * khalf + 1) * 16 + n];
                    c = __builtin_amdgcn_wmma_f32_16x16x4_f32(
                            false, a, false, bb, (short)0, c, false, false);
                }
                // D 16x16 fp32: VGPR v, lanes 0-15 -> row v, lanes 16-31 -> row v+8.
                // Only N-cols 0,1 are live (the two chains).
                if (n < 2) {
                    const int rb = p0 + khalf * 8;
                    float* yo = Yb[n];
                    yo[rb + 0] = c[0]; yo[rb + 1] = c[1];
                    yo[rb + 2] = c[2]; yo[rb + 3] = c[3];
                    yo[rb + 4] = c[4]; yo[rb + 5] = c[5];
                    yo[rb + 6] = c[6]; yo[rb + 7] = c[7];
                }
            }
            __syncthreads();
            if (tid < PDIM) {
                Hb[tid * 16 + 0] = leaky_clip(0.8f * Hb[tid * 16 + 0] + Yb[0][tid]);
                Hb[tid * 16 + 1] = leaky_clip(0.8f * Hb[tid * 16 + 1] + Yb[1][tid]);
            }
            __syncthreads();
        }
        // p_x = Hb[:,0], p_gen = Hb[:,1]

        // g_mem_mu = leaky(clip(p_x @ W_repeat @ W_up))
        if (tid < GDTOT) {
            float acc = 0.0f;
            for (int p = 0; p < PDIM; ++p) acc += Hb[p * 16 + 0] * W_repeat[p * GDTOT + tid];
            tmp40[tid] = acc;
        }
        __syncthreads();
        if (tid < GTOT) {
            float acc = 0.0f;
            for (int d = 0; d < GDTOT; ++d) acc += tmp40[d] * W_up[d * GTOT + tid];
            g_mem_mu[tid] = leaky_clip(acc);
        }
        __syncthreads();
        // g_mem_lv + precision-weighted combine -> g_new (stored into g_vec carry)
        if (tid < GTOT) {
            float acc = b_sig_m[tid];
            for (int i = 0; i < GTOT; ++i) acc += g_mem_mu[i] * W_sig_m[i * GTOT + tid];
            float lv2 = -2.0f + 6.0f * tanhf(acc * (1.0f / 6.0f));
            float p1 = expf(-g_prior_lv[tid]);
            float p2 = expf(-lv2);
            g_vec[tid] = (g_prior_mu[tid] * p1 + g_mem_mu[tid] * p2) / (p1 + p2);
        }
        __syncthreads();
        if (tid < GDTOT) gsel_new[tid] = g_vec[c_gdown[tid]];
        __syncthreads();

        // p_inf = leaky(clip((g_new[gdown] @ W_repeat^T) * px_raw)); d/s vectors for Hebbian
        if (tid < PDIM) {
            const float* wr = W_repeat + tid * GDTOT;
            float acc = 0.0f;
            for (int d = 0; d < GDTOT; ++d) acc += gsel_new[d] * wr[d];
            float pi = leaky_clip(acc * px_raw[tid]);
            float pg = Hb[tid * 16 + 1];
            dvec[tid] = pi - pg;
            svec[tid] = pi + pg;
        }
        __syncthreads();

        // decoder: x_dec = w_x*(p_gen[:100] @ W_tile[:100,:10]) + b_x
        if (tid < SDIM) {
            float acc = 0.0f;
            for (int p = 0; p < 100; ++p) acc += Hb[p * 16 + 1] * W_tile[p * STOT + tid];
            x_dec[tid] = wx * acc + b_x[tid];
        }
        __syncthreads();
        if (tid < HDIM) {
            float acc = b1[tid];
            for (int s = 0; s < SDIM; ++s) acc += x_dec[s] * W1[s * HDIM + tid];
            hid[tid] = acc > 0.0f ? acc : (expf(acc) - 1.0f);
        }
        __syncthreads();
        if (tid < NSENS) {
            float acc = b2[tid];
            for (int j = 0; j < HDIM; ++j) acc += hid[j] * W2[j * NSENS + tid];
            out[((size_t)t * BSIZE + b) * NSENS + tid] = acc;
        }

        // Hebbian RMW: M = clip(0.9999*M + 0.5 * d ⊗ s, -1, 1)  (float4 streaming, L2-resident)
        {
            float4* M4 = (float4*)Mb;
            const int rowq = PDIM / 4;
            for (int i4 = tid; i4 < PDIM * rowq; i4 += NTHREADS) {
                int p  = i4 / rowq;
                int q0 = (i4 - p * rowq) * 4;
                float hdp = 0.5f * dvec[p];
                float4 m = M4[i4];
                m.x = fminf(1.0f, fmaxf(-1.0f, 0.9999f * m.x + hdp * svec[q0 + 0]));
                m.y = fminf(1.0f, fmaxf(-1.0f, 0.9999f * m.y + hdp * svec[q0 + 1]));
                m.z = fminf(1.0f, fmaxf(-1.0f, 0.9999f * m.z + hdp * svec[q0 + 2]));
                m.w = fminf(1.0f, fmaxf(-1.0f, 0.9999f * m.w + hdp * svec[q0 + 3]));
                M4[i4] = m;
            }
        }
        __threadfence_block();
        __syncthreads();
    }
}

extern "C" void kernel_launch(void* const* d_in, const int* in_sizes, int n_in,
                              void* d_out, int out_size, void* d_ws, size_t ws_size,
                              hipStream_t stream) {
    (void)in_sizes; (void)n_in; (void)out_size; (void)ws_size;
    const int*   x_idx    = (const int*)  d_in[0];
    const int*   actions  = (const int*)  d_in[1];
    const float* two_hot  = (const float*)d_in[2];
    const float* W_trans  = (const float*)d_in[3];
    const float* W_sig_p  = (const float*)d_in[4];
    const float* b_sig_p  = (const float*)d_in[5];
    const float* W_up     = (const float*)d_in[6];
    const float* W_sig_m  = (const float*)d_in[7];
    const float* b_sig_m  = (const float*)d_in[8];
    const float* W_repeat = (const float*)d_in[9];
    const float* W_tile   = (const float*)d_in[10];
    const float* w_x      = (const float*)d_in[11];
    const float* b_x      = (const float*)d_in[12];
    const float* W1       = (const float*)d_in[13];
    const float* b1       = (const float*)d_in[14];
    const float* W2       = (const float*)d_in[15];
    const float* b2       = (const float*)d_in[16];

    // Workspace: B * 400*400 fp32 Hebbian matrices = 20.48 MB (L2-resident on MI455X).
    tem_scan_kernel<<<BSIZE, NTHREADS, 0, stream>>>(
        x_idx, actions, two_hot, W_trans, W_sig_p, b_sig_p, W_up, W_sig_m, b_sig_m,
        W_repeat, W_tile, w_x, b_x, W1, b1, W2, b2,
        (float*)d_out, (float*)d_ws);
}